// OutputBlock_62646392979553
// MI455X (gfx1250) — compile-verified
//
#include <hip/hip_runtime.h>

#define EMBED 128
#define LDS_STRIDE 132   // 128 + 4 pad -> conflict-free column-strided LDS reads

typedef float v2f __attribute__((ext_vector_type(2)));
typedef float v8f __attribute__((ext_vector_type(8)));

// ---------------------------------------------------------------------------
// Kernel 1: zero the node accumulator (d_out used in place)
// ---------------------------------------------------------------------------
__global__ void zero_kernel(float4* __restrict__ p, int n4) {
    int i = blockIdx.x * blockDim.x + threadIdx.x;
    if (i < n4) p[i] = make_float4(0.f, 0.f, 0.f, 0.f);
}

// ---------------------------------------------------------------------------
// Kernel 2: edge phase. Grid-stride, one wave per edge iteration,
// 4 channels per lane. Each lane keeps its 32 W_edge coefficients in
// registers so the steady-state traffic is only the m_ji / e_rbf / nbr
// stream (HBM-bound) plus f32 atomics into the 5 MB L2-resident accumulator.
// ---------------------------------------------------------------------------
__global__ __launch_bounds__(256) void edge_kernel(
    const float* __restrict__ m_ji, const float* __restrict__ e_rbf,
    const int* __restrict__ nbr, const float* __restrict__ W_edge,
    float* __restrict__ acc, int n_edges)
{
    const int lane = threadIdx.x & 31;
    const int wid  = (blockIdx.x * 256 + threadIdx.x) >> 5;
    const int nw   = gridDim.x * 8;          // total waves in grid
    const int c0   = lane * 4;

    // Preload this lane's 4 rows of W_edge (4 x 8 floats) into registers.
    float4 w0[4], w1[4];
#pragma unroll
    for (int j = 0; j < 4; ++j) {
        w0[j] = *(const float4*)&W_edge[(size_t)(c0 + j) * 8];
        w1[j] = *(const float4*)&W_edge[(size_t)(c0 + j) * 8 + 4];
    }

    for (int edge = wid; edge < n_edges; edge += nw) {
        const int atom = nbr[edge * 2];
        const float4 r0 = *(const float4*)&e_rbf[(size_t)edge * 8];
        const float4 r1 = *(const float4*)&e_rbf[(size_t)edge * 8 + 4];
        const float4 m  = *(const float4*)&m_ji[(size_t)edge * EMBED + c0];

        // Prefetch next iteration's m_ji segment (global_prefetch_b8 path).
        const int ne = edge + nw;
        if (ne < n_edges)
            __builtin_prefetch(&m_ji[(size_t)ne * EMBED + c0], 0, 1);

        float vals[4];
#pragma unroll
        for (int j = 0; j < 4; ++j) {
            vals[j] = w0[j].x * r0.x + w0[j].y * r0.y +
                      w0[j].z * r0.z + w0[j].w * r0.w +
                      w1[j].x * r1.x + w1[j].y * r1.y +
                      w1[j].z * r1.z + w1[j].w * r1.w;
        }

        float* dst = &acc[(size_t)atom * EMBED + c0];
        unsafeAtomicAdd(&dst[0], vals[0] * m.x);
        unsafeAtomicAdd(&dst[1], vals[1] * m.y);
        unsafeAtomicAdd(&dst[2], vals[2] * m.z);
        unsafeAtomicAdd(&dst[3], vals[3] * m.w);
    }
}

// ---------------------------------------------------------------------------
// Kernel 3: node MLP, fp32 WMMA 16x16x4.
// Block = 256 threads = 8 waves, handles 16 atom rows.
// Wave w owns output columns [16w, 16w+16). K=128 in 32 WMMA steps.
// Fragment layouts per CDNA5 ISA 7.12.2:
//   A 16x4 : lane m = lane&15, VGPRs = {A[m][k], A[m][k+1]}, k base 0/2 per half
//   B 4x16 : lane n = lane&15, VGPRs = {B[k][n], B[k+1][n]}  (B[k][n] == W[n][k])
//   C/D    : VGPR v -> M = v + 8*(lane>=16), N = lane&15
// ---------------------------------------------------------------------------
__global__ __launch_bounds__(256) void node_mlp_kernel(
    const float* __restrict__ acc,
    const float* __restrict__ W1, const float* __restrict__ b1,
    const float* __restrict__ W2, const float* __restrict__ b2,
    const float* __restrict__ W3, const float* __restrict__ b3,
    const float* __restrict__ Wf,
    float* __restrict__ out, int num_atoms)
{
    __shared__ __align__(16) float bufA[16 * LDS_STRIDE];
    __shared__ __align__(16) float bufB[16 * LDS_STRIDE];

    const int tid  = threadIdx.x;
    const int wave = tid >> 5;
    const int lane = tid & 31;
    const int row0 = blockIdx.x * 16;

    // Cooperative load of this block's 16x128 activation tile into LDS.
    for (int idx = tid; idx < 16 * 32; idx += 256) {
        const int r  = idx >> 5;
        const int c4 = idx & 31;
        float4 v = make_float4(0.f, 0.f, 0.f, 0.f);
        if (row0 + r < num_atoms)
            v = *(const float4*)&acc[(size_t)(row0 + r) * EMBED + c4 * 4];
        *(float4*)&bufA[r * LDS_STRIDE + c4 * 4] = v;
    }
    __syncthreads();

    const int n_base = wave * 16;
    const int mN     = lane & 15;       // A row / B col / D col within tile
    const int hi     = lane >> 4;
    const int k_off  = hi * 2;          // K sub-offset per half-wave
    const int mrow   = hi * 8;          // D row base per half-wave

    const float* Ws[3] = {W1, W2, W3};
    const float* bs[3] = {b1, b2, b3};
    float* cur = bufA;
    float* nxt = bufB;

    // Fully unrolled so Ws[layer]/bs[layer] constant-fold to kernel args
    // (keeps the weight loads as GLOBAL_LOAD, not flat).
#pragma unroll
    for (int layer = 0; layer < 3; ++layer) {
        const float* wrow = &Ws[layer][(size_t)(n_base + mN) * EMBED];
        const float  bias = bs[layer][n_base + mN];

        v8f c = {0.f, 0.f, 0.f, 0.f, 0.f, 0.f, 0.f, 0.f};
#pragma unroll 4
        for (int kb = 0; kb < 32; ++kb) {
            const int k = kb * 4 + k_off;
            v2f a = *(const v2f*)&cur[mN * LDS_STRIDE + k];
            v2f b = *(const v2f*)&wrow[k];
            c = __builtin_amdgcn_wmma_f32_16x16x4_f32(
                    false, a, false, b, (short)0, c, false, false);
        }

        // bias + SiLU (fast: x * v_rcp(1 + e^-x)), scatter into other buffer
#pragma unroll
        for (int v = 0; v < 8; ++v) {
            const float x = c[v] + bias;
            const float s = x * __builtin_amdgcn_rcpf(1.0f + __expf(-x));
            nxt[(mrow + v) * LDS_STRIDE + n_base + mN] = s;
        }
        __syncthreads();
        float* t = cur; cur = nxt; nxt = t;
    }

    // Final linear (no bias, no activation) -> global output
    {
        const float* wrow = &Wf[(size_t)(n_base + mN) * EMBED];
        v8f c = {0.f, 0.f, 0.f, 0.f, 0.f, 0.f, 0.f, 0.f};
#pragma unroll 4
        for (int kb = 0; kb < 32; ++kb) {
            const int k = kb * 4 + k_off;
            v2f a = *(const v2f*)&cur[mN * LDS_STRIDE + k];
            v2f b = *(const v2f*)&wrow[k];
            c = __builtin_amdgcn_wmma_f32_16x16x4_f32(
                    false, a, false, b, (short)0, c, false, false);
        }
#pragma unroll
        for (int v = 0; v < 8; ++v) {
            const int row = row0 + mrow + v;
            if (row < num_atoms)
                out[(size_t)row * EMBED + n_base + mN] = c[v];
        }
    }
}

// ---------------------------------------------------------------------------
extern "C" void kernel_launch(void* const* d_in, const int* in_sizes, int n_in,
                              void* d_out, int out_size, void* d_ws, size_t ws_size,
                              hipStream_t stream) {
    const float* m_ji   = (const float*)d_in[0];
    const float* e_rbf  = (const float*)d_in[1];
    const int*   nbr    = (const int*)d_in[2];
    // d_in[3] = num_atoms device scalar (unused; derived from out_size)
    const float* W_edge = (const float*)d_in[4];
    const float* W1 = (const float*)d_in[5];
    const float* B1 = (const float*)d_in[6];
    const float* W2 = (const float*)d_in[7];
    const float* B2 = (const float*)d_in[8];
    const float* W3 = (const float*)d_in[9];
    const float* B3 = (const float*)d_in[10];
    const float* Wf = (const float*)d_in[11];
    float* out = (float*)d_out;

    const int n_edges   = in_sizes[0] / EMBED;
    const int num_atoms = out_size / EMBED;
    const int nfeat     = num_atoms * EMBED;

    // 1) zero accumulator (d_out used in place as the segment-sum target)
    const int n4 = nfeat / 4;
    zero_kernel<<<(n4 + 255) / 256, 256, 0, stream>>>((float4*)out, n4);

    // 2) edge phase: grid-stride, 8 waves per block; cap grid so each wave
    //    amortizes its register-resident W_edge tile over many edges.
    int eblocks = (n_edges + 7) / 8;
    if (eblocks > 4096) eblocks = 4096;
    if (eblocks < 1) eblocks = 1;
    edge_kernel<<<eblocks, 256, 0, stream>>>(
        m_ji, e_rbf, nbr, W_edge, out, n_edges);

    // 3) node MLP with fp32 WMMA, 16 atom rows per block, in place on d_out
    node_mlp_kernel<<<(num_atoms + 15) / 16, 256, 0, stream>>>(
        out, W1, B1, W2, B2, W3, B3, Wf, out, num_atoms);
}